// VQExpert_75076028334464
// MI455X (gfx1250) — compile-verified
//
#include <hip/hip_runtime.h>
#include <stdint.h>

// ---------------------------------------------------------------------------
// VQ-Expert fused persistent kernel for gfx1250 (MI455X).
//   z   = (x @ Wd^T + bd) @ Wpi^T + bpi
//   out = clamp((cb[argmin_c |z-c|^2] @ Wpo^T + bpo) @ Wup^T + bup, -1, 1)
// - VQ distance GEMM (5.2 GFLOP, K=8) on V_WMMA_F32_16X16X4_F32, codebook
//   resident in LDS; score = |c|^2 - 2 z.c (row-constant |z|^2 dropped).
// - Persistent blocks (1 per WGP): codebook/weights staged ONCE per WGP, codebook via
//   global_load_async_to_lds_b32 (ASYNCcnt DMA path, no VGPR roundtrip).
// - Per-wave double-buffered async x-tile staging hides HBM latency behind
//   the 313-tile WMMA search.
// ---------------------------------------------------------------------------

#define N_ROWS      65536
#define IN_F        64
#define HID         32
#define CD          8
#define NCODES      5000
#define NCODES_PAD  5008                 // pad to multiple of 16
#define CTILES      (NCODES_PAD / 16)    // 313
#define NTILES      (N_ROWS / 16)        // 4096 row-tiles
#define WAVES_PER_BLK 8

typedef __attribute__((ext_vector_type(2))) float v2f;
typedef __attribute__((ext_vector_type(8))) float v8f;

// LDS image (bank-conflict padded rows: +1 float where lanes stride by rows).
struct Smem {
  float cb[NCODES_PAD][CD + 1];          // codebook rows (stride 9 -> conflict-free)
  float c2[NCODES_PAD];                  // |c|^2 per code (pad codes = +inf)
  float Wd[HID][IN_F + 1];               // W_down [32][64]
  float bd[HID];
  float Wpi[CD][HID + 1];                // W_pi [8][32]
  float bpi[CD];
  float Wpo[HID][CD + 1];                // W_po [32][8]
  float bpo[HID];
  float Wup[IN_F][HID + 1];              // W_up [64][32]
  float bup[IN_F];
  float xw[WAVES_PER_BLK][2][16][IN_F + 1]; // double-buffered per-wave x tiles
  float ht[WAVES_PER_BLK][16][HID + 1];  // h, later reused for h2
  float zt[WAVES_PER_BLK][16][CD + 1];   // z tile (WMMA A operand source)
  int   it[WAVES_PER_BLK][16];           // argmin index per row
};

extern __shared__ char smem_raw[];

// Async global->LDS copy (CDNA5). LDS address operand = addrspace(3) byte
// offset = low 32 bits of the flat shared pointer (shared aperture occupies
// only the high dword per the ISA aperture check).
__device__ __forceinline__ uint32_t lds_off(const void* p) {
  return (uint32_t)(uintptr_t)p;
}
__device__ __forceinline__ void async_b32(uint32_t lds, uint32_t goff,
                                          const void* sbase) {
  asm volatile("global_load_async_to_lds_b32 %0, %1, %2"
               :: "v"(lds), "v"(goff), "s"(sbase) : "memory");
}
__device__ __forceinline__ void wait_async0()  {
  asm volatile("s_wait_asynccnt 0x0" ::: "memory");
}
__device__ __forceinline__ void wait_async32() {
  asm volatile("s_wait_asynccnt 0x20" ::: "memory");  // prev 32-op batch done
}

__global__ __launch_bounds__(256) void vq_expert_kernel(
    const float* __restrict__ x,
    const float* __restrict__ W_down, const float* __restrict__ b_down,
    const float* __restrict__ W_pi,   const float* __restrict__ b_pi,
    const float* __restrict__ codebook,
    const float* __restrict__ W_po,   const float* __restrict__ b_po,
    const float* __restrict__ W_up,   const float* __restrict__ b_up,
    float* __restrict__ out, int* __restrict__ idx_out,
    float* __restrict__ loss_out)
{
  Smem& s = *reinterpret_cast<Smem*>(smem_raw);
  const int tid  = threadIdx.x;
  const int lane = tid & 31;
  const int w    = tid >> 5;        // wave id in block
  const int half = lane >> 4;       // 0: lanes 0-15, 1: lanes 16-31
  const int l16  = lane & 15;
  const int nwaves = gridDim.x * WAVES_PER_BLK;           // persistent waves
  const int wave0  = blockIdx.x * WAVES_PER_BLK + w;      // this wave's id

  // -------- prefetch this wave's first x tile (overlaps codebook DMA) ------
  if (wave0 < NTILES) {
    for (int i = lane; i < 16 * IN_F; i += 32)
      async_b32(lds_off(&s.xw[w][0][i / IN_F][i % IN_F]),
                (uint32_t)((wave0 * 16 * IN_F + i) * 4), x);
  }
  // -------- codebook -> LDS via async DMA (once per WGP, no VGPR trip) -----
  for (int i = tid; i < NCODES * CD; i += 256)
    async_b32(lds_off(&s.cb[i / CD][i % CD]), (uint32_t)(i * 4), codebook);
  for (int i = NCODES * CD + tid; i < NCODES_PAD * CD; i += 256)
    s.cb[i / CD][i % CD] = 0.0f;
  // -------- |c|^2 and the tiny weight/bias tables (VALU) -------------------
  for (int c = tid; c < NCODES_PAD; c += 256) {
    if (c < NCODES) {
      float a = 0.0f;
      #pragma unroll
      for (int k = 0; k < CD; ++k) { float v = codebook[c * CD + k]; a += v * v; }
      s.c2[c] = a;
    } else {
      s.c2[c] = 3.0e38f;              // padded codes can never win argmin
    }
  }
  for (int i = tid; i < HID * IN_F; i += 256) s.Wd[i / IN_F][i % IN_F] = W_down[i];
  for (int i = tid; i < HID;        i += 256) s.bd[i]  = b_down[i];
  for (int i = tid; i < CD * HID;   i += 256) s.Wpi[i / HID][i % HID]  = W_pi[i];
  for (int i = tid; i < CD;         i += 256) s.bpi[i] = b_pi[i];
  for (int i = tid; i < HID * CD;   i += 256) s.Wpo[i / CD][i % CD]    = W_po[i];
  for (int i = tid; i < HID;        i += 256) s.bpo[i] = b_po[i];
  for (int i = tid; i < IN_F * HID; i += 256) s.Wup[i / HID][i % HID]  = W_up[i];
  for (int i = tid; i < IN_F;       i += 256) s.bup[i] = b_up[i];
  wait_async0();
  __syncthreads();

  // A/B lane layout for V_WMMA_F32_16X16X4_F32:
  //   VGPR0: lanes 0-15 -> K=k0+0, lanes 16-31 -> K=k0+2 (M/N = lane%16)
  //   VGPR1: lanes 0-15 -> K=k0+1, lanes 16-31 -> K=k0+3
  // C/D: VGPR v -> M = v + 8*half, N = lane%16.
  const int ka = (half ? 2 : 0);
  const int kb = (half ? 3 : 1);

  int buf = 0;
  for (int tile = wave0; tile < NTILES; tile += nwaves) {
    // ---- double-buffer: start DMA of next tile, wait for current ----------
    const int nxt = tile + nwaves;
    if (nxt < NTILES) {
      for (int i = lane; i < 16 * IN_F; i += 32)
        async_b32(lds_off(&s.xw[w][buf ^ 1][i / IN_F][i % IN_F]),
                  (uint32_t)((nxt * 16 * IN_F + i) * 4), x);
      wait_async32();                 // in-order: current tile's batch done
    } else {
      wait_async0();
    }
    const int rbase = tile * 16;

    // ---- down-proj: h = x @ Wd^T + bd (WMMA, K=64) ------------------------
    #pragma unroll
    for (int nt = 0; nt < 2; ++nt) {
      v8f acc;
      float bias = s.bd[nt * 16 + l16];
      #pragma unroll
      for (int v = 0; v < 8; ++v) acc[v] = bias;
      #pragma unroll 4
      for (int k0 = 0; k0 < IN_F; k0 += 4) {
        v2f a, b;
        a.x = s.xw[w][buf][l16][k0 + ka];
        a.y = s.xw[w][buf][l16][k0 + kb];
        b.x = s.Wd[nt * 16 + l16][k0 + ka];
        b.y = s.Wd[nt * 16 + l16][k0 + kb];
        acc = __builtin_amdgcn_wmma_f32_16x16x4_f32(false, a, false, b,
                                                    (short)0, acc, false, false);
      }
      #pragma unroll
      for (int v = 0; v < 8; ++v) s.ht[w][v + half * 8][nt * 16 + l16] = acc[v];
    }

    // ---- project-in: z = h @ Wpi^T + bpi (VALU, tiny) ---------------------
    #pragma unroll
    for (int cc = 0; cc < 4; ++cc) {
      int c = half * 4 + cc;
      float a = s.bpi[c];
      #pragma unroll
      for (int j = 0; j < HID; ++j) a += s.ht[w][l16][j] * s.Wpi[c][j];
      s.zt[w][l16][c] = a;
    }

    // ---- VQ search: 313 tiles of 16 codes (WMMA, K=8) ---------------------
    float minv[8]; int mini[8];
    #pragma unroll
    for (int v = 0; v < 8; ++v) { minv[v] = 3.4e38f; mini[v] = 0; }

    for (int t = 0; t < CTILES; ++t) {
      const int cbase = t * 16;
      v8f acc;
      #pragma unroll
      for (int v = 0; v < 8; ++v) acc[v] = 0.0f;
      #pragma unroll
      for (int k0 = 0; k0 < CD; k0 += 4) {
        v2f a, b;
        a.x = s.zt[w][l16][k0 + ka];           // loop-invariant (hoisted)
        a.y = s.zt[w][l16][k0 + kb];
        b.x = s.cb[cbase + l16][k0 + ka];
        b.y = s.cb[cbase + l16][k0 + kb];
        acc = __builtin_amdgcn_wmma_f32_16x16x4_f32(false, a, false, b,
                                                    (short)0, acc, false, false);
      }
      const int   code = cbase + l16;
      const float cc2  = s.c2[code];
      #pragma unroll
      for (int v = 0; v < 8; ++v) {
        float score = cc2 - 2.0f * acc[v];     // |c|^2 - 2 z.c
        if (score < minv[v]) { minv[v] = score; mini[v] = code; }
      }
    }

    // ---- min-reduce over 16 lanes of each half, first-index tie-break -----
    #pragma unroll
    for (int m = 1; m < 16; m <<= 1) {
      #pragma unroll
      for (int v = 0; v < 8; ++v) {
        float ov = __shfl_xor(minv[v], m, 32);
        int   oi = __shfl_xor(mini[v], m, 32);
        if (ov < minv[v] || (ov == minv[v] && oi < mini[v])) {
          minv[v] = ov; mini[v] = oi;
        }
      }
    }
    if (l16 == 0) {
      #pragma unroll
      for (int v = 0; v < 8; ++v) {
        int row = half * 8 + v;
        s.it[w][row] = mini[v];
        idx_out[rbase + row] = mini[v];
      }
    }

    // ---- tail: q -> h2 (per-lane rows) ------------------------------------
    {
      const int code = s.it[w][l16];          // same-wave LDS, in-order
      float q[CD];
      #pragma unroll
      for (int c = 0; c < CD; ++c) q[c] = s.cb[code][c];

      #pragma unroll
      for (int jj = 0; jj < 16; ++jj) {       // h2 = q @ Wpo^T + bpo
        int j = half * 16 + jj;
        float a = s.bpo[j];
        #pragma unroll
        for (int c = 0; c < CD; ++c) a += q[c] * s.Wpo[j][c];
        s.ht[w][l16][j] = a;                  // reuse ht for h2
      }
    }

    // ---- out = clamp(h2 @ Wup^T + bup): coalesced 128B row stores ---------
    {
      const size_t obase = (size_t)rbase * IN_F;
      for (int r2 = 0; r2 < 16; ++r2) {
        float a0 = s.bup[lane];
        float a1 = s.bup[lane + 32];
        #pragma unroll
        for (int j = 0; j < HID; ++j) {
          float h = s.ht[w][r2][j];           // broadcast read, no conflict
          a0 += h * s.Wup[lane][j];
          a1 += h * s.Wup[lane + 32][j];
        }
        out[obase + (size_t)r2 * IN_F + lane]      = fminf(fmaxf(a0, -1.f), 1.f);
        out[obase + (size_t)r2 * IN_F + lane + 32] = fminf(fmaxf(a1, -1.f), 1.f);
      }
    }

    buf ^= 1;
  }

  if (blockIdx.x == 0 && tid == 0) loss_out[0] = 0.0f;   // commit_loss == 0
}

extern "C" void kernel_launch(void* const* d_in, const int* in_sizes, int n_in,
                              void* d_out, int out_size, void* d_ws, size_t ws_size,
                              hipStream_t stream) {
  const float* x    = (const float*)d_in[0];
  const float* Wd   = (const float*)d_in[1];
  const float* bd   = (const float*)d_in[2];
  const float* Wpi  = (const float*)d_in[3];
  const float* bpi  = (const float*)d_in[4];
  const float* cb   = (const float*)d_in[5];
  const float* Wpo  = (const float*)d_in[6];
  const float* bpo  = (const float*)d_in[7];
  const float* Wup  = (const float*)d_in[8];
  const float* bup  = (const float*)d_in[9];

  float* out      = (float*)d_out;
  int*   idx_out  = (int*)(out + (size_t)N_ROWS * IN_F);
  float* loss_out = out + (size_t)N_ROWS * IN_F + N_ROWS;

  // Persistent launch: one block per WGP (multiProcessorCount), capped by the
  // number of 16-row tiles each wave can own. Deterministic, capture-safe.
  int dev = 0, nmp = 0;
  hipGetDevice(&dev);
  hipDeviceGetAttribute(&nmp, hipDeviceAttributeMultiprocessorCount, dev);
  if (nmp <= 0) nmp = 256;
  int grid = nmp < (NTILES / WAVES_PER_BLK) ? nmp : (NTILES / WAVES_PER_BLK);

  const size_t shmem = sizeof(Smem);   // ~301 KB < 320 KB/WGP
  hipFuncSetAttribute(reinterpret_cast<const void*>(vq_expert_kernel),
                      hipFuncAttributeMaxDynamicSharedMemorySize, (int)shmem);

  vq_expert_kernel<<<grid, 256, shmem, stream>>>(
      x, Wd, bd, Wpi, bpi, cb, Wpo, bpo, Wup, bup, out, idx_out, loss_out);
}